// SynthesisNetwork_43121471652452
// MI455X (gfx1250) — compile-verified
//
#include <hip/hip_runtime.h>
#include <math.h>

typedef __attribute__((ext_vector_type(2))) float v2f;
typedef __attribute__((ext_vector_type(8))) float v8f;

#define SQRT2F 1.41421356237309515f

__device__ __forceinline__ float actf(float x, int act) {
  // act 0 = ELU, 1 = leaky-relu(0.2)
  return act == 0 ? (x > 0.f ? x : (__expf(x) - 1.f))
                  : (x > 0.f ? x : 0.2f * x);
}

// ---------------- broadcast const (1,512,16) -> (8,512,16) ----------------
__global__ void k_bcast(const float* __restrict__ c, float* __restrict__ f) {
  int t = blockIdx.x * blockDim.x + threadIdx.x;
  if (t < 8 * 8192) f[t] = c[t & 8191];
}

__global__ void k_zero(float* __restrict__ p, int n) {
  int t = blockIdx.x * blockDim.x + threadIdx.x;
  if (t < n) p[t] = 0.f;
}

// ---------------- style affine: s[b,c] = (w[b,:]·aff_w[c,:])/sqrt(512)+aff_b[c]
__global__ void k_styles(const float* __restrict__ w, const float* __restrict__ aw,
                         const float* __restrict__ ab, float* __restrict__ s, int C) {
  int t = blockIdx.x * blockDim.x + threadIdx.x;
  if (t >= 8 * C) return;
  int b = t / C, c = t % C;
  const float* wr = w + b * 512;
  const float* ar = aw + c * 512;
  float acc = 0.f;
  for (int j = 0; j < 512; ++j) acc += wr[j] * ar[j];
  s[t] = acc * 0.044194173824159216f + ab[c];
}

// ---------------- demod: d[b,o] = rsqrt(sum_ik (Wc*sc)^2 s^2 + 1e-8) --------
__global__ void k_demod(const float* __restrict__ cw, const float* __restrict__ s,
                        float* __restrict__ d, int Ci, int Co) {
  int t = blockIdx.x * blockDim.x + threadIdx.x;
  if (t >= 8 * Co) return;
  int b = t / Co, o = t % Co;
  const float* wr = cw + (size_t)o * Ci * 3;
  const float* sr = s + b * Ci;
  float acc = 0.f;
  for (int i = 0; i < Ci; ++i) {
    float w0 = wr[3 * i], w1 = wr[3 * i + 1], w2 = wr[3 * i + 2];
    float si = sr[i];
    acc += (w0 * w0 + w1 * w1 + w2 * w2) * si * si;
  }
  float sc2 = 1.f / (3.f * (float)Ci);
  d[t] = rsqrtf(acc * sc2 + 1e-8f);
}

// ---------------- modulated conv k=3 pad=1, + noise/bias/amp/act ------------
__global__ void k_modconv(const float* __restrict__ fin, const float* __restrict__ cw,
                          const float* __restrict__ s, const float* __restrict__ d,
                          const float* __restrict__ noise, const float* __restrict__ bias,
                          const float* __restrict__ nscale, float* __restrict__ fout,
                          int Ci, int Co, int W, int act) {
  long t = (long)blockIdx.x * blockDim.x + threadIdx.x;
  if (t >= (long)8 * Co * W) return;
  int x = (int)(t % W);
  int o = (int)((t / W) % Co);
  int b = (int)(t / ((long)W * Co));
  float acc = 0.f;
  for (int i = 0; i < Ci; ++i) {
    float si = s[b * Ci + i];
    const float* fr = fin + ((size_t)b * Ci + i) * W;
    const float* wr = cw + ((size_t)o * Ci + i) * 3;
    if (x > 0)     acc += wr[0] * si * fr[x - 1];
    acc += wr[1] * si * fr[x];
    if (x < W - 1) acc += wr[2] * si * fr[x + 1];
  }
  float y = acc * rsqrtf(3.f * (float)Ci) * d[b * Co + o];
  y += nscale[0] * noise[x];
  y = (y + bias[o]) * SQRT2F;
  fout[t] = actf(y, act);
}

// ---------------- transposed conv stride 2, VALID: Win -> 2*Win+1 -----------
__global__ void k_upconv_pre(const float* __restrict__ fin, const float* __restrict__ cw,
                             const float* __restrict__ s, const float* __restrict__ d,
                             float* __restrict__ pre, int Ci, int Co, int Win) {
  int Wp = 2 * Win + 1;
  long t = (long)blockIdx.x * blockDim.x + threadIdx.x;
  if (t >= (long)8 * Co * Wp) return;
  int x = (int)(t % Wp);
  int o = (int)((t / Wp) % Co);
  int b = (int)(t / ((long)Wp * Co));
  float acc = 0.f;
  for (int i = 0; i < Ci; ++i) {
    float si = s[b * Ci + i];
    const float* fr = fin + ((size_t)b * Ci + i) * Win;
    const float* wr = cw + ((size_t)o * Ci + i) * 3;
    for (int k = 0; k < 3; ++k) {
      int tx = x - k;
      if (tx >= 0 && !(tx & 1)) {
        int xx = tx >> 1;
        if (xx < Win) acc += wr[k] * si * fr[xx];
      }
    }
  }
  pre[t] = acc * rsqrtf(3.f * (float)Ci) * d[b * Co + o];
}

// ---------------- blur [1,3,3,1]/8 pad1 (Wp -> Wp-1=2Win) + noise/bias/act --
__global__ void k_blur_post(const float* __restrict__ pre, const float* __restrict__ noise,
                            const float* __restrict__ bias, const float* __restrict__ nscale,
                            float* __restrict__ fout, int Co, int Wout, int act) {
  int Wp = Wout + 1;
  long t = (long)blockIdx.x * blockDim.x + threadIdx.x;
  if (t >= (long)8 * Co * Wout) return;
  int x = (int)(t % Wout);
  int o = (int)((t / Wout) % Co);
  int b = (int)(t / ((long)Wout * Co));
  const float* pr = pre + ((size_t)b * Co + o) * Wp;
  float acc = 0.375f * pr[x] + 0.375f * pr[x + 1];
  if (x >= 1)     acc += 0.125f * pr[x - 1];
  if (x + 2 < Wp) acc += 0.125f * pr[x + 2];
  float y = acc + nscale[0] * noise[x];
  y = (y + bias[o]) * SQRT2F;
  fout[t] = actf(y, act);
}

// ---------------- to_wave: wv[b,x] = sum_i s2*cw/sqrt(C)*fmap + b, skip-sum -
__global__ void k_towave(const float* __restrict__ fmap, const float* __restrict__ s2,
                         const float* __restrict__ cw, const float* __restrict__ bias,
                         const float* __restrict__ prev, float* __restrict__ outw,
                         int C, int W) {
  long t = (long)blockIdx.x * blockDim.x + threadIdx.x;
  if (t >= (long)8 * W) return;
  int x = (int)(t % W);
  int b = (int)(t / W);
  float acc = 0.f;
  for (int i = 0; i < C; ++i)
    acc += s2[b * C + i] * cw[i] * fmap[((size_t)b * C + i) * W + x];
  acc = acc * rsqrtf((float)C) + bias[0];
  if (prev) acc += prev[(size_t)b * (W >> 1) + (x >> 1)];
  outw[t] = acc;
}

// ---------------- label GEMM via V_WMMA_F32_16X16X4_F32 ---------------------
// Ypart[8,N] += X[8,K-chunk] * (W[N,K]^T / sqrt(K))   (atomic K-split combine)
// A (activations, pre-scaled by 1/sqrt(K)) staged in LDS: rows 8..15 zeroed,
// so fragment reads are unconditional ds_load_b64 (no EXEC juggling).
// One wave = one 16x16 tile; gridDim.y = K-split for HBM-latency hiding.
#define LBK 256
__global__ void __launch_bounds__(128) k_label_wmma(
    const float* __restrict__ X, const float* __restrict__ Wt,
    float* __restrict__ Ypart, int K, int N, int kchunk) {
  __shared__ float As[16 * LBK];

  int lane = threadIdx.x & 31;
  int wave = threadIdx.x >> 5;
  int col0 = (blockIdx.x * 4 + wave) * 16;

  int mr  = lane & 15;            // A row (batch)
  int ko  = (lane >> 4) << 1;     // K sub-offset {0,2}
  int col = col0 + (lane & 15);   // B/D column
  int cc  = col < N ? col : N - 1;

  const float* wrow = Wt + (size_t)cc * K;
  float wsc = rsqrtf((float)K);

  int k_begin = blockIdx.y * kchunk;
  int k_end   = k_begin + kchunk;   // kchunk divides K exactly (host)

  // zero pad rows 8..15 once (stay zero for the whole kernel)
  for (int j = threadIdx.x; j < 8 * LBK; j += blockDim.x) As[8 * LBK + j] = 0.f;

  v8f acc = {0.f, 0.f, 0.f, 0.f, 0.f, 0.f, 0.f, 0.f};
  for (int kb = k_begin; kb < k_end; kb += LBK) {
    int kcur = k_end - kb;
    if (kcur > LBK) kcur = LBK;
    // cooperative coalesced staging of A rows 0..7, pre-scaled
    for (int j = threadIdx.x; j < 8 * LBK; j += blockDim.x) {
      int r = j >> 8, kk = j & (LBK - 1);
      As[j] = (kk < kcur) ? X[(size_t)r * K + kb + kk] * wsc : 0.f;
    }
    __syncthreads();
#pragma unroll 4
    for (int k0 = 0; k0 < kcur; k0 += 4) {
      v2f a, bf;
      a.x  = As[(mr << 8) + k0 + ko];
      a.y  = As[(mr << 8) + k0 + ko + 1];
      bf.x = wrow[kb + k0 + ko];
      bf.y = wrow[kb + k0 + ko + 1];
      __builtin_prefetch(wrow + kb + k0 + 512, 0, 1);  // global_prefetch_b8
      acc = __builtin_amdgcn_wmma_f32_16x16x4_f32(
          false, a, false, bf, (short)0, acc, false, false);
    }
    __syncthreads();
  }

  if (lane < 16 && col < N) {     // lanes 0-15 hold batch rows 0-7 (v = M)
#pragma unroll
    for (int v = 0; v < 8; ++v)
      atomicAdd(&Ypart[(size_t)v * N + col], acc[v]);
  }
}

// bias + optional leaky-relu epilogue
__global__ void k_bias_act(const float* __restrict__ Ypart, const float* __restrict__ bias,
                           float* __restrict__ out, int N, int act) {
  int t = blockIdx.x * blockDim.x + threadIdx.x;
  if (t >= 8 * N) return;
  float r = Ypart[t] + bias[t % N];
  if (act) r = r > 0.f ? r : 0.2f * r;
  out[t] = r;
}

// ===========================================================================
extern "C" void kernel_launch(void* const* d_in, const int* in_sizes, int n_in,
                              void* d_out, int out_size, void* d_ws, size_t ws_size,
                              hipStream_t stream) {
  // JAX pytree (sorted-dict) flattening order:
  // params.const, params.label[(W,b)x8], params.stages[{aff_b,aff_w,bias,conv_w,noise,noise_scale}x19],
  // params.to_wave[{aff_b,aff_w,bias,conv_w}x10], w
  int idx = 0;
  auto F = [&](int i) { return (const float*)d_in[i]; };
  const float* Pconst = F(idx++);
  const float *LW[8], *LB[8];
  for (int i = 0; i < 8; ++i) { LW[i] = F(idx++); LB[i] = F(idx++); }
  const float *st_ab[19], *st_aw[19], *st_bias[19], *st_cw[19], *st_noise[19], *st_ns[19];
  for (int t = 0; t < 19; ++t) {
    st_ab[t] = F(idx++); st_aw[t] = F(idx++); st_bias[t] = F(idx++);
    st_cw[t] = F(idx++); st_noise[t] = F(idx++); st_ns[t] = F(idx++);
  }
  const float *tw_ab[10], *tw_aw[10], *tw_bias[10], *tw_cw[10];
  for (int t = 0; t < 10; ++t) {
    tw_ab[t] = F(idx++); tw_aw[t] = F(idx++); tw_bias[t] = F(idx++); tw_cw[t] = F(idx++);
  }
  const float* Wlat = F(idx++);

  // workspace carve-up (floats)
  float* ws = (float*)d_ws;
  const size_t FM = (size_t)8 * 34816;       // >= 8*512*65 and 8*4*8193
  float* fmA = ws;
  float* fmB = fmA + FM;
  float* pre = fmB + FM;
  float* S   = pre + FM;
  float* Dd  = S  + 8 * 512;
  float* S2  = Dd + 8 * 512;
  float* wvA = S2 + 8 * 512;
  float* wvB = wvA + 8 * 8192;
  float* lxP = wvB + 8 * 8192;    // GEMM partial accumulator
  float* lxA = lxP + 8 * 8192;    // layer output ping
  float* lxB = lxA + 8 * 8192;    // layer output pong

  static const int kindA[19] = {0,1,0,1,0,1,0,1,0,1,0,1,0,1,0,1,0,1,0};
  static const int ciA[19]   = {512,512,512,512,512,512,256,256,128,128,64,64,32,32,16,16,8,8,4};
  static const int coA[19]   = {512,512,512,512,512,256,256,128,128,64,64,32,32,16,16,8,8,4,4};
  static const int woA[19]   = {16,32,32,64,64,128,128,256,256,512,512,1024,1024,2048,2048,4096,4096,8192,8192};

  auto gsz = [](long n, int bs) { return (unsigned)((n + bs - 1) / bs); };

  k_bcast<<<gsz(8 * 8192, 256), 256, 0, stream>>>(Pconst, fmA);

  float* cur = fmA;
  float* nxt = fmB;
  const float* wave_prev = nullptr;
  float* wave_bufs[2] = {wvA, wvB};
  int wtog = 0, tw = 0;

  for (int t = 0; t < 19; ++t) {
    int Ci = ciA[t], Co = coA[t], Wout = woA[t];
    int act = (t >= 17) ? 1 : 0;
    k_styles<<<gsz(8 * Ci, 128), 128, 0, stream>>>(Wlat, st_aw[t], st_ab[t], S, Ci);
    k_demod<<<gsz(8 * Co, 128), 128, 0, stream>>>(st_cw[t], S, Dd, Ci, Co);
    if (kindA[t] == 1) {
      int Win = Wout / 2;
      k_upconv_pre<<<gsz((long)8 * Co * (2 * Win + 1), 256), 256, 0, stream>>>(
          cur, st_cw[t], S, Dd, pre, Ci, Co, Win);
      k_blur_post<<<gsz((long)8 * Co * Wout, 256), 256, 0, stream>>>(
          pre, st_noise[t], st_bias[t], st_ns[t], nxt, Co, Wout, act);
    } else {
      k_modconv<<<gsz((long)8 * Co * Wout, 256), 256, 0, stream>>>(
          cur, st_cw[t], S, Dd, st_noise[t], st_bias[t], st_ns[t], nxt, Ci, Co, Wout, act);
    }
    float* tmp = cur; cur = nxt; nxt = tmp;
    if (kindA[t] == 0) {
      k_styles<<<gsz(8 * Co, 128), 128, 0, stream>>>(Wlat, tw_aw[tw], tw_ab[tw], S2, Co);
      float* outw = (tw == 9) ? (float*)d_out : wave_bufs[wtog];
      k_towave<<<gsz((long)8 * Wout, 256), 256, 0, stream>>>(
          cur, S2, tw_cw[tw], tw_bias[tw], wave_prev, outw, Co, Wout);
      wave_prev = outw; wtog ^= 1; ++tw;
    }
  }

  // label network: final fmap (8,4,8192) is already contiguous (8,32768)
  static const int LD[9] = {32768, 8192, 2048, 512, 256, 128, 64, 32, 6};
  const float* x = cur;
  float* lbufs[2] = {lxA, lxB};
  for (int i = 0; i < 8; ++i) {
    int K = LD[i], N = LD[i + 1];
    // K-split: keep each wave's chunk ~4K so layer 0 spawns 4096 waves
    int ksplit = (K >= 32768) ? 8 : (K >= 8192) ? 4 : (K >= 2048) ? 2 : 1;
    int kchunk = K / ksplit;                 // K, ksplit both powers of two
    int ntiles = (N + 15) / 16;
    k_zero<<<gsz(8 * N, 256), 256, 0, stream>>>(lxP, 8 * N);
    k_label_wmma<<<dim3((unsigned)((ntiles + 3) / 4), (unsigned)ksplit), 128, 0, stream>>>(
        x, LW[i], lxP, K, N, kchunk);
    float* out = (i == 7) ? ((float*)d_out + 8 * 8192) : lbufs[i & 1];
    k_bias_act<<<gsz(8 * N, 256), 256, 0, stream>>>(lxP, LB[i], out, N, (i < 7) ? 1 : 0);
    x = out;
  }
}